// BasicGCN_42717744726283
// MI455X (gfx1250) — compile-verified
//
#include <hip/hip_runtime.h>
#include <hip/hip_bf16.h>

// ---------------------------------------------------------------------------
// BasicGCN on gfx1250 (MI455X).  GEMMs via V_WMMA_F32_16X16X4_F32 (exact fp32)
// with weights pre-packed into the B-fragment layout (branch-free inner loop).
// Message aggregation via per-call CSR build (no fp32 atomics).
// ---------------------------------------------------------------------------

typedef __attribute__((ext_vector_type(2))) float v2f;
typedef __attribute__((ext_vector_type(8))) float v8f;

#define HIDDEN 256
#define IN_DIM 84          // 7*12
#define OUT_DIM 168        // 7*24

// ---------------------------------------------------------------------------
// small helper kernels
// ---------------------------------------------------------------------------
__global__ void k_init_node(float* __restrict__ deg, int* __restrict__ counts, int n) {
    int i = blockIdx.x * blockDim.x + threadIdx.x;
    if (i < n) { deg[i] = 1.0f; counts[i] = 0; }   // self-loop weight 1
}

__global__ void k_deg_count(const int* __restrict__ dst, const float* __restrict__ ew,
                            float* __restrict__ deg, int* __restrict__ counts, int E) {
    int e = blockIdx.x * blockDim.x + threadIdx.x;
    if (e < E) {
        int d = dst[e];
        atomicAdd(&deg[d], ew[e]);
        atomicAdd(&counts[d], 1);
    }
}

__global__ void k_dinv(const float* __restrict__ deg, float* __restrict__ dinv,
                       float* __restrict__ selfnorm, int n) {
    int i = blockIdx.x * blockDim.x + threadIdx.x;
    if (i < n) {
        float dv = deg[i];
        float r = (dv > 0.0f) ? rsqrtf(dv) : 0.0f;
        dinv[i] = r;
        selfnorm[i] = r * r;          // self-loop: dinv[i]*1*dinv[i]
    }
}

// single-block exclusive scan over counts[n] -> row_ptr[n+1], cursor copy
__global__ __launch_bounds__(1024) void k_scan(const int* __restrict__ counts,
                                               int* __restrict__ row_ptr,
                                               int* __restrict__ cursor, int n) {
    __shared__ int tmp[1024];
    __shared__ int carry;
    if (threadIdx.x == 0) carry = 0;
    __syncthreads();
    int nChunks = (n + 1023) / 1024;
    for (int c = 0; c < nChunks; ++c) {
        int i = c * 1024 + threadIdx.x;
        int v = (i < n) ? counts[i] : 0;
        tmp[threadIdx.x] = v;
        __syncthreads();
        for (int off = 1; off < 1024; off <<= 1) {
            int t = (threadIdx.x >= off) ? tmp[threadIdx.x - off] : 0;
            __syncthreads();
            tmp[threadIdx.x] += t;
            __syncthreads();
        }
        int incl = tmp[threadIdx.x];
        int excl = incl - v + carry;
        if (i < n) { row_ptr[i] = excl; cursor[i] = excl; }
        __syncthreads();
        if (threadIdx.x == 1023) carry += tmp[1023];
        __syncthreads();
    }
    if (threadIdx.x == 0) row_ptr[n] = carry;
}

__global__ void k_csr_fill(const int* __restrict__ src, const int* __restrict__ dst,
                           const float* __restrict__ ew, const float* __restrict__ dinv,
                           int* __restrict__ cursor, int* __restrict__ col_src,
                           float* __restrict__ enorm, int E) {
    int e = blockIdx.x * blockDim.x + threadIdx.x;
    if (e < E) {
        int s = src[e];
        int d = dst[e];
        int pos = atomicAdd(&cursor[d], 1);
        col_src[pos] = s;
        enorm[pos] = dinv[s] * ew[e] * dinv[d];
    }
}

// ---------------------------------------------------------------------------
// Pack W[K,N] into WMMA B-fragment order.  One thread per (ks, nt, lane):
//   Wp[((ks*ntiles + nt)*32 + lane)*2 + {0,1}] = W[ks*4 + half*2 + {0,1}][nt*16 + l15]
// Invalid columns (col >= N) are zero-filled, so the GEMM needs no guards.
// ---------------------------------------------------------------------------
__global__ void k_pack_w(const float* __restrict__ W, float* __restrict__ Wp,
                         int N, int K) {
    const int ntiles = (N + 15) >> 4;
    const int total = (K >> 2) * ntiles * 32;
    int idx = blockIdx.x * blockDim.x + threadIdx.x;
    if (idx >= total) return;
    int lane = idx & 31;
    int nt   = (idx >> 5) % ntiles;
    int ks   = idx / (32 * ntiles);
    int half = lane >> 4;
    int l15  = lane & 15;
    int ka   = ks * 4 + half * 2;
    int col  = nt * 16 + l15;
    float b0 = 0.0f, b1 = 0.0f;
    if (col < N) {
        b0 = W[(size_t)ka * N + col];
        b1 = W[(size_t)(ka + 1) * N + col];
    }
    Wp[(size_t)idx * 2 + 0] = b0;
    Wp[(size_t)idx * 2 + 1] = b1;
}

// ---------------------------------------------------------------------------
// WMMA fp32 GEMM:  C[M,N] = act(A[M,K]) * W[K,N] (+ bias[N]), W pre-packed.
// block = 256 threads = 8 waves; block handles a 16-row strip, wave w owns
// N-tiles {w, w+8}.  Branch-free inner loop, one b64 load per fragment.
// Requires M % 16 == 0, K % 4 == 0.
// ---------------------------------------------------------------------------
template <int RELU_IN>
__global__ __launch_bounds__(256)
void k_gemm_wmma_f32(const float* __restrict__ A, const float* __restrict__ Wp,
                     const float* __restrict__ bias, float* __restrict__ C,
                     int N, int K) {
    const int lane = threadIdx.x & 31;
    const int wave = __builtin_amdgcn_readfirstlane(threadIdx.x >> 5);  // 0..7 scalar
    const int half = lane >> 4;
    const int l15  = lane & 15;
    const int m0   = blockIdx.x * 16;
    const int ntiles = (N + 15) >> 4;
    const int nsteps = K >> 2;

    const int nt0 = wave;
    const int nt1 = wave + 8;
    const bool has1 = (nt1 < ntiles);                 // scalar condition

    // A fragment: lanes 0-15 -> rows m0..m0+15, K={k,k+1}; lanes 16-31 K={k+2,k+3}
    const float* __restrict__ Arow = A + (size_t)(m0 + l15) * K + half * 2;
    const float* __restrict__ B0   = Wp + ((size_t)nt0 * 32 + lane) * 2;
    const float* __restrict__ B1   = Wp + ((size_t)nt1 * 32 + lane) * 2;
    const size_t bstride = (size_t)ntiles * 64;       // floats per k-step

    v8f acc0 = {};
    v8f acc1 = {};

    if (has1) {
#pragma unroll 4
        for (int ks = 0; ks < nsteps; ++ks) {
            v2f a = *(const v2f*)(Arow + 4 * ks);
            if (RELU_IN) { a.x = fmaxf(a.x, 0.0f); a.y = fmaxf(a.y, 0.0f); }
            v2f b0 = *(const v2f*)(B0 + ks * bstride);
            v2f b1 = *(const v2f*)(B1 + ks * bstride);
            acc0 = __builtin_amdgcn_wmma_f32_16x16x4_f32(
                false, a, false, b0, (short)0, acc0, false, false);
            acc1 = __builtin_amdgcn_wmma_f32_16x16x4_f32(
                false, a, false, b1, (short)0, acc1, false, false);
        }
    } else {
#pragma unroll 4
        for (int ks = 0; ks < nsteps; ++ks) {
            v2f a = *(const v2f*)(Arow + 4 * ks);
            if (RELU_IN) { a.x = fmaxf(a.x, 0.0f); a.y = fmaxf(a.y, 0.0f); }
            v2f b0 = *(const v2f*)(B0 + ks * bstride);
            acc0 = __builtin_amdgcn_wmma_f32_16x16x4_f32(
                false, a, false, b0, (short)0, acc0, false, false);
        }
    }

    // C/D layout: VGPR j -> row m0 + j + (half ? 8 : 0), col = tile*16 + l15
    const int col0 = nt0 * 16 + l15;
    if (col0 < N) {
        float bv = bias ? bias[col0] : 0.0f;
#pragma unroll
        for (int j = 0; j < 8; ++j) {
            int row = m0 + j + half * 8;
            C[(size_t)row * N + col0] = acc0[j] + bv;
        }
    }
    const int col1 = nt1 * 16 + l15;
    if (has1 && col1 < N) {
        float bv = bias ? bias[col1] : 0.0f;
#pragma unroll
        for (int j = 0; j < 8; ++j) {
            int row = m0 + j + half * 8;
            C[(size_t)row * N + col1] = acc1[j] + bv;
        }
    }
}

// ---------------------------------------------------------------------------
// Aggregate: Hout[i,c] = selfnorm[i]*T[i,c] + bias[c] + sum_e enorm[e]*T[src_e,c]
// one 256-thread block per node, thread = channel.  ReLU deferred to next read.
// ---------------------------------------------------------------------------
__global__ __launch_bounds__(256)
void k_aggregate(const float* __restrict__ T, const int* __restrict__ row_ptr,
                 const int* __restrict__ col_src, const float* __restrict__ enorm,
                 const float* __restrict__ selfnorm, const float* __restrict__ bias,
                 float* __restrict__ Hout) {
    const int i = blockIdx.x;
    const int c = threadIdx.x;
    float acc = selfnorm[i] * T[(size_t)i * HIDDEN + c] + bias[c];
    const int beg = row_ptr[i];
    const int end = row_ptr[i + 1];
    for (int j = beg; j < end; ++j) {
        int s = col_src[j];
        float w = enorm[j];
        acc += w * T[(size_t)s * HIDDEN + c];
    }
    Hout[(size_t)i * HIDDEN + c] = acc;
}

// ---------------------------------------------------------------------------
// launch
// ---------------------------------------------------------------------------
extern "C" void kernel_launch(void* const* d_in, const int* in_sizes, int n_in,
                              void* d_out, int out_size, void* d_ws, size_t ws_size,
                              hipStream_t stream) {
    const float* x      = (const float*)d_in[0];   // [n, 7, 12] -> [n, 84]
    const int*   ei     = (const int*)d_in[1];     // [2, E]
    const float* ew     = (const float*)d_in[2];   // [E]
    const float* W_emb  = (const float*)d_in[3];   // [84, 256]
    const float* b_emb  = (const float*)d_in[4];   // [256]
    const float* Wg     = (const float*)d_in[5];   // [5, 256, 256]
    const float* bg     = (const float*)d_in[6];   // [5, 256]
    const float* W_dec  = (const float*)d_in[7];   // [256, 168]
    const float* b_dec  = (const float*)d_in[8];   // [168]
    float*       out    = (float*)d_out;           // [n, 168]

    const int n = in_sizes[0] / IN_DIM;            // 50000
    const int E = in_sizes[2];                     // 800000
    const int* src = ei;
    const int* dst = ei + E;

    // ---- workspace carve-out -------------------------------------------------
    float* p = (float*)d_ws;
    float* bufA     = p; p += (size_t)n * HIDDEN;  // h (ping)
    float* bufB     = p; p += (size_t)n * HIDDEN;  // t (pong)
    float* Wp       = p; p += (size_t)65536;       // packed weights (max 256KB)
    float* deg      = p; p += n;
    float* dinv     = p; p += n;
    float* selfnorm = p; p += n;
    float* enorm    = p; p += E;
    int* q = (int*)p;
    int* counts  = q; q += n;
    int* row_ptr = q; q += n + 1;
    int* cursor  = q; q += n;
    int* col_src = q; q += E;

    const int B = 256;
    const int gN = (n + B - 1) / B;
    const int gE = (E + B - 1) / B;
    const int gM = n / 16;                         // 50000 / 16 = 3125

    // ---- normalization + CSR build ------------------------------------------
    k_init_node<<<gN, B, 0, stream>>>(deg, counts, n);
    k_deg_count<<<gE, B, 0, stream>>>(dst, ew, deg, counts, E);
    k_dinv<<<gN, B, 0, stream>>>(deg, dinv, selfnorm, n);
    k_scan<<<1, 1024, 0, stream>>>(counts, row_ptr, cursor, n);
    k_csr_fill<<<gE, B, 0, stream>>>(src, dst, ew, dinv, cursor, col_src, enorm, E);

    // helper for pack grid size
    auto packGrid = [](int N_, int K_) {
        int total = (K_ >> 2) * (((N_ + 15) >> 4)) * 32;
        return (total + 255) / 256;
    };

    // ---- embedding: bufA = x @ W_emb + b_emb  (relu deferred to next read) --
    k_pack_w<<<packGrid(HIDDEN, IN_DIM), B, 0, stream>>>(W_emb, Wp, HIDDEN, IN_DIM);
    k_gemm_wmma_f32<0><<<gM, B, 0, stream>>>(x, Wp, b_emb, bufA, HIDDEN, IN_DIM);

    // ---- 5 GCN layers --------------------------------------------------------
    for (int l = 0; l < 5; ++l) {
        k_pack_w<<<packGrid(HIDDEN, HIDDEN), B, 0, stream>>>(
            Wg + (size_t)l * HIDDEN * HIDDEN, Wp, HIDDEN, HIDDEN);
        // bufB = relu(bufA) @ Wg[l]
        k_gemm_wmma_f32<1><<<gM, B, 0, stream>>>(bufA, Wp, nullptr, bufB,
                                                 HIDDEN, HIDDEN);
        // bufA = aggregate(bufB) + bg[l]   (relu deferred)
        k_aggregate<<<n, B, 0, stream>>>(bufB, row_ptr, col_src, enorm,
                                         selfnorm, bg + (size_t)l * HIDDEN, bufA);
    }

    // ---- decoder: out = relu(bufA) @ W_dec + b_dec --------------------------
    k_pack_w<<<packGrid(OUT_DIM, HIDDEN), B, 0, stream>>>(W_dec, Wp, OUT_DIM, HIDDEN);
    k_gemm_wmma_f32<1><<<gM, B, 0, stream>>>(bufA, Wp, b_dec, out, OUT_DIM, HIDDEN);
}